// CausalFlowModel_51290499449000
// MI455X (gfx1250) — compile-verified
//
#include <hip/hip_runtime.h>
#include <hip/hip_bf16.h>

typedef float v2f __attribute__((ext_vector_type(2)));
typedef float v8f __attribute__((ext_vector_type(8)));

#define B_   2048
#define T_   512
#define SD_  8
#define CD1  9       // CD + 1
#define H_   72      // 64 + SD
#define G4   288     // 4*H
#define KP   84      // K padded: 72 (h) + 9 (x_t) + 3 zero pad -> 21 k-steps of 4
#define NKS  21      // KP/4
#define GS   292     // gate LDS row stride (8*292 % 64 == 32 -> conflict-free D stores)
#define NT   288     // lstm block threads (9 waves)

// ---------------- Kernel 1: encoder MLP -> h0_stack = [x | h0] ----------------
__global__ __launch_bounds__(256) void enc_kernel(
    const float* __restrict__ x,
    const float* __restrict__ W1, const float* __restrict__ b1,
    const float* __restrict__ W2, const float* __restrict__ b2,
    const float* __restrict__ W3, const float* __restrict__ b3,
    float* __restrict__ h0s)
{
  __shared__ float xs[SD_], z1[256], z2[256];
  const int b = blockIdx.x, tid = threadIdx.x;
  if (tid < SD_) xs[tid] = x[b * SD_ + tid];
  __syncthreads();
  {
    float a = b1[tid];
    #pragma unroll
    for (int k = 0; k < SD_; k++) a += W1[tid * SD_ + k] * xs[k];
    z1[tid] = tanhf(a);
  }
  __syncthreads();
  {
    float a = b2[tid];
    for (int k = 0; k < 256; k++) a += W2[tid * 256 + k] * z1[k];
    z2[tid] = tanhf(a);
  }
  __syncthreads();
  if (tid < 64) {
    float a = b3[tid];
    for (int k = 0; k < 256; k++) a += W3[tid * 256 + k] * z2[k];
    h0s[(size_t)b * H_ + SD_ + tid] = a;
  }
  if (tid < SD_) h0s[(size_t)b * H_ + tid] = xs[tid];
}

// Combined weight Wc = [W_hh | W_ih | 0] (288 x 84), read transposed: Wc[n][k]
__device__ __forceinline__ float wval(const float* __restrict__ Whh,
                                      const float* __restrict__ Wih,
                                      int n, int k) {
  if (k < H_)       return Whh[n * H_ + k];
  if (k < H_ + CD1) return Wih[n * CD1 + (k - H_)];
  return 0.f;
}

// ---------------- Kernel 2: persistent-LDS LSTM recurrence (WMMA f32 16x16x4) ----------------
// One block = 16 batch rows, 9 waves; each wave owns 2 of the 18 gate column tiles.
// B (weight) fragments live in VGPRs for the entire T loop.
__global__ __launch_bounds__(NT) void lstm_kernel(
    const float* __restrict__ rnn,      // (B, T, 9)
    const float* __restrict__ deltas,   // (B, T)
    const int*   __restrict__ h_lens,   // (B,)
    const float* __restrict__ W_ih,     // (288, 9)
    const float* __restrict__ W_hh,     // (288, 72)
    const float* __restrict__ b_ih, const float* __restrict__ b_hh,
    const float* __restrict__ h0s,      // (B, 72)
    float* __restrict__ dec_in)         // (B, 72)
{
  __shared__ float S[16][KP];     // A operand: [h_{t-1} | x_t | 0]
  __shared__ float G[16][GS];     // gate pre-activations
  __shared__ float Cc[16][H_];    // cell state
  __shared__ float Hsv[16][H_];   // h snapshot at t == L-2
  __shared__ float Dd[16][H_];    // dec_in accumulator
  __shared__ float bsum[G4];
  __shared__ int   lens[16];
  __shared__ int   maxL;

  const int tid  = threadIdx.x;
  const int row0 = blockIdx.x * 16;

  for (int idx = tid; idx < G4; idx += NT) bsum[idx] = b_ih[idx] + b_hh[idx];
  for (int idx = tid; idx < 16 * KP; idx += NT) {
    int r = idx / KP, j = idx % KP;
    float v = 0.f;
    if (j < H_)            v = h0s[(size_t)(row0 + r) * H_ + j];
    else if (j < H_ + CD1) v = rnn[(size_t)(row0 + r) * T_ * CD1 + (j - H_)];
    S[r][j] = v;
  }
  for (int idx = tid; idx < 16 * H_; idx += NT) Cc[idx / H_][idx % H_] = 0.f;
  if (tid < 16) lens[tid] = h_lens[row0 + tid];
  __syncthreads();
  if (tid == 0) {
    int m = 0;
    for (int r = 0; r < 16; r++) m = max(m, lens[r]);
    maxL = m;
  }

  const int lane = tid & 31;
  const int wv   = tid >> 5;          // wave id 0..8
  const int cb   = wv * 32;           // 2 column tiles of 16
  const int mn   = lane & 15;         // M (for A) / N (for B, D)
  const int koff = (lane >> 4) * 2;   // K sub-offset per ISA A/B layout
  const int drow = (lane >> 4) * 8;   // D-row offset per ISA C/D layout

  // Preload loop-invariant B fragments into registers (one-time, from global).
  v2f breg0[NKS], breg1[NKS];
  #pragma unroll
  for (int ks = 0; ks < NKS; ks++) {
    const int k  = ks * 4 + koff;
    const int n0 = cb + mn, n1 = cb + 16 + mn;
    breg0[ks].x = wval(W_hh, W_ih, n0, k);
    breg0[ks].y = wval(W_hh, W_ih, n0, k + 1);
    breg1[ks].x = wval(W_hh, W_ih, n1, k);
    breg1[ks].y = wval(W_hh, W_ih, n1, k + 1);
  }
  __syncthreads();
  const int nsteps = maxL;

  const bool xth = (tid < 16 * CD1);
  const int  xr  = tid / CD1, xk = tid % CD1;

  for (int t = 0; t < nsteps; t++) {
    // Prefetch x_{t+1} (independent of this step's compute).
    float xnext = 0.f;
    const bool doX = xth && (t + 1 < T_);
    if (doX) xnext = rnn[(size_t)(row0 + xr) * T_ * CD1 + (size_t)(t + 1) * CD1 + xk];

    v8f acc0 = {}, acc1 = {};
    #pragma unroll
    for (int ks = 0; ks < NKS; ks++) {
      const int k = ks * 4 + koff;
      v2f av; av.x = S[mn][k]; av.y = S[mn][k + 1];
      acc0 = __builtin_amdgcn_wmma_f32_16x16x4_f32(false, av, false, breg0[ks], (short)0, acc0, false, false);
      acc1 = __builtin_amdgcn_wmma_f32_16x16x4_f32(false, av, false, breg1[ks], (short)0, acc1, false, false);
    }
    #pragma unroll
    for (int v = 0; v < 8; v++) {
      G[drow + v][cb + mn]      = acc0[v];
      G[drow + v][cb + 16 + mn] = acc1[v];
    }
    __syncthreads();

    // Commit prefetched x_{t+1} (WMMA reads of S are done).
    if (doX) S[xr][H_ + xk] = xnext;

    // Elementwise LSTM cell update: 16*72 = 1152 elements, 4 per thread.
    #pragma unroll
    for (int e = 0; e < 4; e++) {
      const int idx = tid * 4 + e;
      const int r = idx / H_, j = idx % H_;
      float gi = G[r][j]          + bsum[j];
      float gf = G[r][H_ + j]     + bsum[H_ + j];
      float gg = G[r][2 * H_ + j] + bsum[2 * H_ + j];
      float go = G[r][3 * H_ + j] + bsum[3 * H_ + j];
      float i_ = 1.f / (1.f + __expf(-gi));
      float f_ = 1.f / (1.f + __expf(-gf));
      float o_ = 1.f / (1.f + __expf(-go));
      float g_ = tanhf(gg);
      float c  = f_ * Cc[r][j] + i_ * g_;
      Cc[r][j] = c;
      float h  = o_ * tanhf(c);
      S[r][j]  = h;
      const int L = lens[r];
      if (t == L - 2) {
        Hsv[r][j] = h;
      } else if (t == L - 1) {
        float d = deltas[(size_t)(row0 + r) * T_ + (L - 1)];
        Dd[r][j] = (1.f - d) * Hsv[r][j] + d * h;
      }
    }
    __syncthreads();
  }

  for (int idx = tid; idx < 16 * H_; idx += NT) {
    const int r = idx / H_, j = idx % H_;
    dec_in[(size_t)(row0 + r) * H_ + j] = Dd[r][j];
  }
}

// ---------------- Kernel 3: decoder MLP ----------------
__global__ __launch_bounds__(288) void dec_kernel(
    const float* __restrict__ din,
    const float* __restrict__ W1, const float* __restrict__ b1,
    const float* __restrict__ W2, const float* __restrict__ b2,
    const float* __restrict__ W3, const float* __restrict__ b3,
    float* __restrict__ out)
{
  __shared__ float vv[H_], z1[288], z2[288];
  const int b = blockIdx.x, tid = threadIdx.x;
  if (tid < H_) vv[tid] = din[(size_t)b * H_ + tid];
  __syncthreads();
  {
    float a = b1[tid];
    for (int k = 0; k < H_; k++) a += W1[tid * H_ + k] * vv[k];
    z1[tid] = tanhf(a);
  }
  __syncthreads();
  {
    float a = b2[tid];
    for (int k = 0; k < 288; k++) a += W2[tid * 288 + k] * z1[k];
    z2[tid] = tanhf(a);
  }
  __syncthreads();
  if (tid < SD_) {
    float a = b3[tid];
    for (int k = 0; k < 288; k++) a += W3[tid * 288 + k] * z2[k];
    out[(size_t)b * SD_ + tid] = a;
  }
}

extern "C" void kernel_launch(void* const* d_in, const int* in_sizes, int n_in,
                              void* d_out, int out_size, void* d_ws, size_t ws_size,
                              hipStream_t stream) {
  const float* x      = (const float*)d_in[0];
  const float* rnn    = (const float*)d_in[1];
  const float* deltas = (const float*)d_in[2];
  const int*   h_lens = (const int*)  d_in[3];
  const float* W_ih   = (const float*)d_in[4];
  const float* W_hh   = (const float*)d_in[5];
  const float* b_ih   = (const float*)d_in[6];
  const float* b_hh   = (const float*)d_in[7];
  const float* eW1 = (const float*)d_in[8];
  const float* eb1 = (const float*)d_in[9];
  const float* eW2 = (const float*)d_in[10];
  const float* eb2 = (const float*)d_in[11];
  const float* eW3 = (const float*)d_in[12];
  const float* eb3 = (const float*)d_in[13];
  const float* dW1 = (const float*)d_in[14];
  const float* db1 = (const float*)d_in[15];
  const float* dW2 = (const float*)d_in[16];
  const float* db2 = (const float*)d_in[17];
  const float* dW3 = (const float*)d_in[18];
  const float* db3 = (const float*)d_in[19];

  float* out    = (float*)d_out;
  float* h0s    = (float*)d_ws;                    // (B, 72)
  float* dec_in = h0s + (size_t)B_ * H_;           // (B, 72)

  enc_kernel <<<B_,      256, 0, stream>>>(x, eW1, eb1, eW2, eb2, eW3, eb3, h0s);
  lstm_kernel<<<B_ / 16, NT,  0, stream>>>(rnn, deltas, h_lens, W_ih, W_hh, b_ih, b_hh, h0s, dec_in);
  dec_kernel <<<B_,      288, 0, stream>>>(dec_in, dW1, db1, dW2, db2, dW3, db3, out);
}